// cut_patches_periodic_padding_23398981829311
// MI455X (gfx1250) — compile-verified
//
#include <hip/hip_runtime.h>

// Pure gather kernel: cut 16x16 periodic patches from images + emit flat indices.
// Memory-movement bound (192 MB of streaming writes); no FLOPs -> no WMMA.
// CDNA5 paths used: async global->LDS staging (ASYNCcnt) + non-temporal b128 stores.

typedef float v4f __attribute__((ext_vector_type(4)));
typedef int   v4i __attribute__((vector_size(16)));   // matches builtin's expected pointee

#define IMG_H   128
#define IMG_W   128
#define IMG_C   3
#define PATCH   16
#define N_IMGS_ 64
#define N_PROJ_ 512
#define HWP     (IMG_H * IMG_W)          // 16384 floats per channel plane
#define CHW     (IMG_C * HWP)            // 49152 floats per image
#define TOTAL   (N_PROJ_ * (long long)N_IMGS_ * IMG_C * PATCH * PATCH) // 25,165,824

#define TPB           192                // 6 wave32s
#define ROWS          (IMG_C * PATCH)    // 48 staged rows per block
#define CHUNKS_PER_ROW 5                 // 5 x b128 = 20 floats covers any wrapped 16-float window
#define LROW          24                 // floats per LDS row slot (20 used + pad)
#define NCHUNK        (ROWS * CHUNKS_PER_ROW) // 240

#if defined(__gfx1250__) && __has_builtin(__builtin_amdgcn_global_load_async_to_lds_b128)
#define USE_ASYNC_LDS 1
#else
#define USE_ASYNC_LDS 0
#endif

__global__ __launch_bounds__(TPB)
void cut_patches_periodic_kernel(const float* __restrict__ imgs,
                                 const int*   __restrict__ pos_h,
                                 const int*   __restrict__ pos_w,
                                 float*       __restrict__ out_patches,
                                 float*       __restrict__ out_inds)
{
    __shared__ float tile[ROWS * LROW];  // 4.6 KB -> high occupancy on 320 KB WGP

    const int t    = threadIdx.x;
    const int blk  = blockIdx.x;              // proj * 64 + n
    const int n    = blk & (N_IMGS_ - 1);
    const int proj = blk >> 6;

    const int posh     = pos_h[proj];         // block-uniform -> SMEM
    const int posw     = pos_w[proj];
    const int s4       = posw & ~3;           // 16B-aligned window start column
    const int img_base = n * CHW;

    // ---- stage: 48 rows x 5 aligned b128 chunks (20-float wrapped window) ----
    for (int i = t; i < NCHUNK; i += TPB) {
        const int r   = i / CHUNKS_PER_ROW;          // 0..47  (= c*16 + ph)
        const int m   = i - r * CHUNKS_PER_ROW;      // 0..4
        const int c   = r >> 4;
        const int ph  = r & (PATCH - 1);
        const int hh  = (posh + ph) & (IMG_H - 1);
        const int col = (s4 + (m << 2)) & (IMG_W - 1);   // chunk start, never straddles row
        const float* gp = imgs + img_base + c * HWP + hh * IMG_W + col;
        float*       lp = &tile[r * LROW + (m << 2)];
#if USE_ASYNC_LDS
        __builtin_amdgcn_global_load_async_to_lds_b128((v4i*)gp, (v4i*)lp, 0, 0);
#else
        *(v4f*)lp = *(const v4f*)gp;
#endif
    }
#if USE_ASYNC_LDS
    asm volatile("s_wait_asynccnt 0" ::: "memory");
#endif
    __syncthreads();

    // ---- emit: each thread writes one float4 of patch data + one float4 of indices ----
    const int c      = t >> 6;                // 0..2
    const int f      = t & 63;
    const int ph     = f >> 2;                // 0..15
    const int pwb    = (f & 3) << 2;          // 0,4,8,12
    const int base_d = (posw & 3) + pwb;      // offset of pw window inside staged 20-float slot
    const float* lrow   = &tile[(c * PATCH + ph) * LROW];
    const int hh        = (posh + ph) & (IMG_H - 1);
    const int lin_row   = img_base + c * HWP + hh * IMG_W;

    v4f pv, iv;
#pragma unroll
    for (int j = 0; j < 4; ++j) {
        const int wj = (posw + pwb + j) & (IMG_W - 1);
        pv[j] = lrow[base_d + j];
        iv[j] = (float)(lin_row + wj);        // gather index < 2^24 -> exact in f32
    }

    const long long obase = (long long)blk * (IMG_C * PATCH * PATCH)
                          + (c << 8) + (ph << 4) + pwb;
    // Streaming 192 MB total: non-temporal so the 12 MB input stays L2-resident.
    __builtin_nontemporal_store(pv, (v4f*)(out_patches + obase));
    __builtin_nontemporal_store(iv, (v4f*)(out_inds + obase));
}

extern "C" void kernel_launch(void* const* d_in, const int* in_sizes, int n_in,
                              void* d_out, int out_size, void* d_ws, size_t ws_size,
                              hipStream_t stream) {
    const float* imgs = (const float*)d_in[0];
    const int*   ph   = (const int*)d_in[1];
    const int*   pw   = (const int*)d_in[2];
    float* outp = (float*)d_out;
    float* outi = outp + (size_t)TOTAL;       // tuple output: patches then linear_inds

    (void)in_sizes; (void)n_in; (void)out_size; (void)d_ws; (void)ws_size;

    cut_patches_periodic_kernel<<<N_PROJ_ * N_IMGS_, TPB, 0, stream>>>(imgs, ph, pw, outp, outi);
}